// GINEncoder_23871428231480
// MI455X (gfx1250) — compile-verified
//
#include <hip/hip_runtime.h>
#include <hip/hip_bf16.h>

typedef float v2f __attribute__((ext_vector_type(2)));
typedef float v8f __attribute__((ext_vector_type(8)));

#define HID 64
#define LDS_STRIDE 68  // 68 % 64 banks = 4 -> conflict-free column reloads

// ---------------- x = node_emb[z] ----------------
__global__ void __launch_bounds__(256) gather_kernel(
    const float* __restrict__ emb, const int* __restrict__ z,
    float* __restrict__ x, int n) {
  int gid = blockIdx.x * blockDim.x + threadIdx.x;
  int i = gid >> 4;
  if (i >= n) return;
  int c = (gid & 15) << 2;
  float4 v = *(const float4*)(emb + (size_t)z[i] * HID + c);
  *(float4*)(x + (size_t)i * HID + c) = v;
}

// ---------------- agg = 0 ----------------
__global__ void __launch_bounds__(256) zero_kernel(float* __restrict__ p, int n4) {
  int gid = blockIdx.x * blockDim.x + threadIdx.x;
  if (gid >= n4) return;
  float4 zv = make_float4(0.f, 0.f, 0.f, 0.f);
  *(float4*)(p + (size_t)gid * 4) = zv;
}

// ---------------- msg = relu(x[src] + ea); agg[dst] += msg ----------------
__global__ void __launch_bounds__(256) edge_kernel(
    const float* __restrict__ x, const float* __restrict__ ea,
    const int* __restrict__ src, const int* __restrict__ dst,
    float* __restrict__ agg, int E) {
  int gid = blockIdx.x * blockDim.x + threadIdx.x;
  int e = gid >> 4;  // 16 threads per edge, float4 each
  if (e >= E) return;
  int c = (gid & 15) << 2;
  int s = src[e];
  int d = dst[e];
  float4 xv = *(const float4*)(x + (size_t)s * HID + c);
  float4 av = *(const float4*)(ea + (size_t)e * HID + c);
  float4 m;
  m.x = fmaxf(xv.x + av.x, 0.f);
  m.y = fmaxf(xv.y + av.y, 0.f);
  m.z = fmaxf(xv.z + av.z, 0.f);
  m.w = fmaxf(xv.w + av.w, 0.f);
  float* ap = agg + (size_t)d * HID + c;
  unsafeAtomicAdd(ap + 0, m.x);   // global_atomic_add_f32
  unsafeAtomicAdd(ap + 1, m.y);
  unsafeAtomicAdd(ap + 2, m.z);
  unsafeAtomicAdd(ap + 3, m.w);
}

// ---------------- fused GINE node MLP via FP32 WMMA ----------------
// One wave32 per 16-node tile.
// out = agg + x ; h = relu(out @ W1^T + b1) ; h = h @ W2^T + b2 ;
// (relu) ; h += x
__global__ void __launch_bounds__(32) mlp_kernel(
    const float* __restrict__ x, const float* __restrict__ agg,
    const float* __restrict__ W1, const float* __restrict__ b1,
    const float* __restrict__ W2, const float* __restrict__ b2,
    float* __restrict__ out, int n, int do_relu) {
  __shared__ float lds[16 * LDS_STRIDE];

  const int lane = threadIdx.x;     // 0..31
  const int r = lane & 15;          // row within tile (A / B col)
  const int g = lane >> 4;          // lane-group (selects K pair)
  const int base = blockIdx.x * 16; // first node of tile
  if (base >= n) return;

  int rrow = base + r;
  if (rrow >= n) rrow = n - 1;      // clamped reads; stores guarded below

  // --- A fragments of (x + agg), ISA 16x4 f32 A layout:
  //     lane l<16: M=l, K = 4*kc + {0,1}; lane l>=16: M=l-16, K = 4*kc + {2,3}
  v2f afr[16];
  {
    const float* xr = x   + (size_t)rrow * HID;
    const float* gr = agg + (size_t)rrow * HID;
#pragma unroll
    for (int kc = 0; kc < 16; ++kc) {
      int col = kc * 4 + g * 2;
      v2f xv = *(const v2f*)(xr + col);
      v2f gv = *(const v2f*)(gr + col);
      afr[kc] = xv + gv;
    }
  }

  const v8f vzero = {0.f, 0.f, 0.f, 0.f, 0.f, 0.f, 0.f, 0.f};
  v8f acc[4];

  // --- GEMM1: h1 = out @ W1^T  (B[k][o] = W1[o][k]) ---
#pragma unroll
  for (int ot = 0; ot < 4; ++ot) acc[ot] = vzero;
#pragma unroll
  for (int kc = 0; kc < 16; ++kc) {
#pragma unroll
    for (int ot = 0; ot < 4; ++ot) {
      v2f b = *(const v2f*)(W1 + (size_t)(ot * 16 + r) * HID + kc * 4 + g * 2);
      acc[ot] = __builtin_amdgcn_wmma_f32_16x16x4_f32(
          false, afr[kc], false, b, (short)0, acc[ot], false, false);
    }
  }

  // --- bias + relu, C/D layout (M = v + 8*g_hi, N = lane%16) -> LDS ---
#pragma unroll
  for (int ot = 0; ot < 4; ++ot) {
    int col = ot * 16 + r;
    float bias = b1[col];
#pragma unroll
    for (int v = 0; v < 8; ++v) {
      float h = acc[ot][v] + bias;
      h = fmaxf(h, 0.f);
      lds[(v + 8 * g) * LDS_STRIDE + col] = h;
    }
  }
  __syncthreads();  // block == one wave: uniform, just orders LDS

  // --- reload h1 as A fragments ---
#pragma unroll
  for (int kc = 0; kc < 16; ++kc) {
    int col = kc * 4 + g * 2;
    afr[kc] = *(const v2f*)(&lds[r * LDS_STRIDE + col]);
  }

  // --- GEMM2: h2 = h1 @ W2^T ---
#pragma unroll
  for (int ot = 0; ot < 4; ++ot) acc[ot] = vzero;
#pragma unroll
  for (int kc = 0; kc < 16; ++kc) {
#pragma unroll
    for (int ot = 0; ot < 4; ++ot) {
      v2f b = *(const v2f*)(W2 + (size_t)(ot * 16 + r) * HID + kc * 4 + g * 2);
      acc[ot] = __builtin_amdgcn_wmma_f32_16x16x4_f32(
          false, afr[kc], false, b, (short)0, acc[ot], false, false);
    }
  }

  // --- epilogue: + b2 ; optional relu ; + x residual ; store ---
#pragma unroll
  for (int ot = 0; ot < 4; ++ot) {
    int col = ot * 16 + r;
    float bias = b2[col];
#pragma unroll
    for (int v = 0; v < 8; ++v) {
      int row = base + v + 8 * g;
      if (row < n) {
        float h = acc[ot][v] + bias;
        if (do_relu) h = fmaxf(h, 0.f);
        h += x[(size_t)row * HID + col];
        out[(size_t)row * HID + col] = h;
      }
    }
  }
}

extern "C" void kernel_launch(void* const* d_in, const int* in_sizes, int n_in,
                              void* d_out, int out_size, void* d_ws, size_t ws_size,
                              hipStream_t stream) {
  const float* node_emb = (const float*)d_in[0];
  const float* W1       = (const float*)d_in[1];
  const float* b1       = (const float*)d_in[2];
  const float* W2       = (const float*)d_in[3];
  const float* b2       = (const float*)d_in[4];
  const float* edge_attr= (const float*)d_in[5];
  const int*   z        = (const int*)d_in[6];
  const int*   eidx     = (const int*)d_in[7];

  const int n = in_sizes[6];
  const int E = in_sizes[7] / 2;
  const int* src = eidx;       // edge_index[0]
  const int* dst = eidx + E;   // edge_index[1]
  float* out = (float*)d_out;

  const size_t xelems = (size_t)n * HID;
  float* xA  = (float*)d_ws;
  float* xB  = xA + xelems;
  float* agg = xB + xelems;

  // x = node_emb[z]
  {
    int total = n * 16;
    gather_kernel<<<(total + 255) / 256, 256, 0, stream>>>(node_emb, z, xA, n);
  }

  const int ntiles = (n + 15) / 16;
  const int n4 = (int)(xelems / 4);
  const float* cur = xA;
  float* nxt = xB;

  for (int conv = 0; conv < 3; ++conv) {
    zero_kernel<<<(n4 + 255) / 256, 256, 0, stream>>>(agg, n4);

    {
      long long etotal = (long long)E * 16;
      edge_kernel<<<(unsigned)((etotal + 255) / 256), 256, 0, stream>>>(
          cur, edge_attr, src, dst, agg, E);
    }

    float* dbuf = (conv == 2) ? out : nxt;
    mlp_kernel<<<ntiles, 32, 0, stream>>>(
        cur, agg,
        W1 + (size_t)conv * HID * HID, b1 + (size_t)conv * HID,
        W2 + (size_t)conv * HID * HID, b2 + (size_t)conv * HID,
        dbuf, n, (conv < 2) ? 1 : 0);

    nxt = (float*)cur;  // old input becomes next scratch
    cur = dbuf;
  }
}